// GINlayer_6665789243400
// MI455X (gfx1250) — compile-verified
//
#include <hip/hip_runtime.h>
#include <hip/hip_bf16.h>

typedef float v2f __attribute__((ext_vector_type(2)));
typedef float v8f __attribute__((ext_vector_type(8)));

#define N_NODES 50000
#define N_EDGES 800000
#define DIN 128
#define DHID 512
#define DOUT 128

// ---------------------------------------------------------------------------
// 1) zero the scatter accumulator (float4-vectorized)
// ---------------------------------------------------------------------------
__global__ __launch_bounds__(256) void gin_zero_kernel(float4* __restrict__ p, int n4) {
    int i = blockIdx.x * blockDim.x + threadIdx.x;
    if (i < n4) p[i] = make_float4(0.f, 0.f, 0.f, 0.f);
}

// ---------------------------------------------------------------------------
// 2) edge gather + scatter-add: nz[dst] += nh[src]
//    one wave32 per edge; each lane moves 4 contiguous floats (coalesced 512B)
// ---------------------------------------------------------------------------
__global__ __launch_bounds__(256) void gin_scatter_kernel(
    const float* __restrict__ nh, const long long* __restrict__ edge_index,
    float* __restrict__ nz, int n_edges)
{
    int edge = (blockIdx.x * blockDim.x + threadIdx.x) >> 5;
    int lane = threadIdx.x & 31;
    if (edge >= n_edges) return;
    long long src = edge_index[edge];            // row 0: src
    long long dst = edge_index[n_edges + edge];  // row 1: dst
    const float4 v = ((const float4*)(nh + src * DIN))[lane];
    float* d = nz + dst * DIN + lane * 4;
    atomicAdd(d + 0, v.x);
    atomicAdd(d + 1, v.y);
    atomicAdd(d + 2, v.z);
    atomicAdd(d + 3, v.w);
}

// ---------------------------------------------------------------------------
// 3) X = (1+eps)*nh + nz  (in-place over nz)
// ---------------------------------------------------------------------------
__global__ __launch_bounds__(256) void gin_residual_kernel(
    const float* __restrict__ nh, const float* __restrict__ eps,
    float* __restrict__ nz, int n)
{
    int i = blockIdx.x * blockDim.x + threadIdx.x;
    if (i < n) {
        float s = 1.0f + eps[0];
        nz[i] = fmaf(s, nh[i], nz[i]);
    }
}

// ---------------------------------------------------------------------------
// 3b) weight transpose: out[c*R + r] = in[r*C + c]
//     lets the WMMA B-operand (two consecutive K at fixed N) become one b64
// ---------------------------------------------------------------------------
__global__ __launch_bounds__(256) void gin_transpose_kernel(
    const float* __restrict__ in, float* __restrict__ out, int R, int C)
{
    int idx = blockIdx.x * blockDim.x + threadIdx.x;
    if (idx < R * C) {
        int r = idx / C, c = idx % C;
        out[(size_t)c * R + r] = in[idx];
    }
}

// ---------------------------------------------------------------------------
// 4) fused GIN MLP: out = relu(X*W1 + b1)*W2 + b2, 16 rows per block.
//    Hidden activation (16x512 f32, 32KB) lives entirely in LDS.
//    X tile staged via GLOBAL_LOAD_ASYNC_TO_LDS_B128 (ASYNCcnt path).
//    Weights pre-transposed: W1t[n][k] (512x128), W2t[n][k] (128x512) so the
//    B operand (rows kk+2h, kk+2h+1 at col n0+l) is one aligned b64 load.
//    WMMA: V_WMMA_F32_16X16X4_F32, fp32 in / fp32 accumulate.
// ---------------------------------------------------------------------------
__global__ __launch_bounds__(256) void gin_mlp_kernel(
    const float* __restrict__ X,   const float* __restrict__ W1t,
    const float* __restrict__ b1,  const float* __restrict__ W2t,
    const float* __restrict__ b2,  float* __restrict__ out)
{
    // single allocation => Xs sits at LDS offset 0 (needed by the async asm)
    __shared__ float smem[16 * DIN + 16 * DHID];   // 8 KB + 32 KB
    float* Xs = smem;
    float* Hs = smem + 16 * DIN;

    const int rowBase = blockIdx.x * 16;
    const int tid  = threadIdx.x;     // 0..255
    const int wave = tid >> 5;        // 0..7
    const int lane = tid & 31;
    const int half = lane >> 4;       // ISA A/B layout: lanes 16-31 carry K+2
    const int l    = lane & 15;

    // ---- async DMA the 16x128 X tile into LDS (2 x b128 per thread) ----
    // INST_OFFSET is added to BOTH the LDS and global addresses, so the same
    // register pair covers both 4 KB passes.
    {
        unsigned ldsoff = (unsigned)(tid * 16);                    // byte offset in LDS
        const float* gptr = X + (size_t)rowBase * DIN + tid * 4;   // 16B per thread
        asm volatile(
            "global_load_async_to_lds_b128 %0, %1, off\n\t"
            "global_load_async_to_lds_b128 %0, %1, off offset:4096\n\t"
            "s_wait_asynccnt 0"
            :: "v"(ldsoff), "v"(gptr) : "memory");
    }
    __syncthreads();

    // ---- GEMM1 + bias + ReLU -> LDS ----
    #pragma unroll
    for (int t = 0; t < 4; ++t) {
        const int n0 = (wave * 4 + t) * 16;
        v8f c = {};
        #pragma unroll 4
        for (int kk = 0; kk < DIN; kk += 4) {
            // A (16x4): lane 16h+l holds X[row=l][kk+2h], X[row=l][kk+2h+1]
            v2f a = *(const v2f*)(Xs + l * DIN + kk + 2 * half);
            // B (4x16): lane 16h+l holds W1[kk+2h][n0+l], W1[kk+2h+1][n0+l]
            v2f b = *(const v2f*)(W1t + (size_t)(n0 + l) * DIN + kk + 2 * half);
            c = __builtin_amdgcn_wmma_f32_16x16x4_f32(
                    false, a, false, b, (short)0, c, false, false);
        }
        const float bias = b1[n0 + l];
        #pragma unroll
        for (int i = 0; i < 8; ++i) {
            // D VGPR i in lane 16h+l = (M=i+8h, N=n0+l)
            float v = c[i] + bias;
            Hs[(i + 8 * half) * DHID + n0 + l] = v > 0.f ? v : 0.f;
        }
    }
    __syncthreads();

    // ---- GEMM2 + bias -> global ----
    {
        const int n0 = wave * 16;   // 8 waves x 16 = 128 output cols
        v8f c = {};
        #pragma unroll 4
        for (int kk = 0; kk < DHID; kk += 4) {
            v2f a = *(const v2f*)(Hs + l * DHID + kk + 2 * half);
            v2f b = *(const v2f*)(W2t + (size_t)(n0 + l) * DHID + kk + 2 * half);
            c = __builtin_amdgcn_wmma_f32_16x16x4_f32(
                    false, a, false, b, (short)0, c, false, false);
        }
        const float bias = b2[n0 + l];
        #pragma unroll
        for (int i = 0; i < 8; ++i) {
            out[(size_t)(rowBase + i + 8 * half) * DOUT + n0 + l] = c[i] + bias;
        }
    }
}

// ---------------------------------------------------------------------------
// Launch: inputs in setup_inputs() order:
//   0 nh, 1 eh, 2 W1, 3 b1, 4 W2, 5 b2, 6 eps, 7 edge_index
// Output: [ n_h (50000x128) | eh (800000x128) ] flat f32
// Workspace: [ nz 25.6MB | W1t 256KB | W2t 256KB ]
// ---------------------------------------------------------------------------
extern "C" void kernel_launch(void* const* d_in, const int* in_sizes, int n_in,
                              void* d_out, int out_size, void* d_ws, size_t ws_size,
                              hipStream_t stream) {
    const float*     nh  = (const float*)d_in[0];
    const float*     eh  = (const float*)d_in[1];
    const float*     W1  = (const float*)d_in[2];
    const float*     b1  = (const float*)d_in[3];
    const float*     W2  = (const float*)d_in[4];
    const float*     b2  = (const float*)d_in[5];
    const float*     eps = (const float*)d_in[6];
    const long long* ei  = (const long long*)d_in[7];  // int64 per reference

    const int NZ  = N_NODES * DIN;             // 6,400,000
    const int NZ4 = NZ / 4;
    const int NW  = DIN * DHID;                // 65,536 weights per matrix

    float* nz  = (float*)d_ws;
    float* W1t = nz + NZ;                      // 512 x 128 (row = hidden col)
    float* W2t = W1t + NW;                     // 128 x 512 (row = out col)
    float* out = (float*)d_out;

    // 1) zero accumulator (must happen every call; ws is not re-poisoned)
    gin_zero_kernel<<<(NZ4 + 255) / 256, 256, 0, stream>>>((float4*)nz, NZ4);

    // 2) scatter-add over edges: one wave per edge
    {
        int threads_total = N_EDGES * 32;
        gin_scatter_kernel<<<(threads_total + 255) / 256, 256, 0, stream>>>(
            nh, ei, nz, N_EDGES);
    }

    // 3) epsilon residual (in-place)
    gin_residual_kernel<<<(NZ + 255) / 256, 256, 0, stream>>>(nh, eps, nz, NZ);

    // 3b) weight transposes (tiny: 256 KB each, once per call)
    gin_transpose_kernel<<<(NW + 255) / 256, 256, 0, stream>>>(W1, W1t, DIN, DHID);
    gin_transpose_kernel<<<(NW + 255) / 256, 256, 0, stream>>>(W2, W2t, DHID, DOUT);

    // 4) fused WMMA MLP -> first output slot
    gin_mlp_kernel<<<N_NODES / 16, 256, 0, stream>>>(nz, W1t, b1, W2t, b2, out);

    // 5) eh pass-through -> second output slot
    hipMemcpyAsync(out + (size_t)N_NODES * DOUT, eh,
                   (size_t)N_EDGES * DIN * sizeof(float),
                   hipMemcpyDeviceToDevice, stream);
}